// PointNet_33569464385759
// MI455X (gfx1250) — compile-verified
//
#include <hip/hip_runtime.h>
#include <hip/hip_bf16.h>

// ---------------------------------------------------------------------------
// PointNet++ semantic-segmentation forward for MI455X (gfx1250, wave32).
// All MLP layers run on V_WMMA_F32_16X16X4_F32 (full f32 precision, matching
// the f32 reference). SA stages fuse gather + 3 MLP layers + maxpool per
// ball-group (16 samples == one WMMA M-tile) with LDS-resident activations.
// Weights with K % 4 != 0 are pre-packed (zero-padded) so the WMMA K-loops
// are branch-free b64 loads; BatchNorm is pre-folded to scale/shift arrays.
// ---------------------------------------------------------------------------

typedef float v2f __attribute__((ext_vector_type(2)));
typedef float v8f __attribute__((ext_vector_type(8)));

__device__ __forceinline__ v8f wmma4(v2f a, v2f b, v8f c) {
  // D(16x16,f32) = A(16x4,f32) * B(4x16,f32) + C
  return __builtin_amdgcn_wmma_f32_16x16x4_f32(false, a, false, b, (short)0, c,
                                               false, false);
}

// ---------------------------------------------------------------- split input
__global__ void k_split(const float* __restrict__ pc, float* __restrict__ xyz,
                        float* __restrict__ feat, int BN) {
  int i = blockIdx.x * blockDim.x + threadIdx.x;
  if (i >= BN) return;
  const float4 v = ((const float4*)pc)[i];
  xyz[i * 3 + 0] = v.x;
  xyz[i * 3 + 1] = v.y;
  xyz[i * 3 + 2] = v.z;
  feat[i] = v.w;
}

// ----------------------------------------------------- weight pad / BN fold
__global__ void k_pack_w(const float* __restrict__ W, float* __restrict__ Wp,
                         int O, int K, int Kp) {
  int i = blockIdx.x * blockDim.x + threadIdx.x;
  if (i >= O * Kp) return;
  int o = i / Kp, c = i % Kp;
  Wp[i] = (c < K) ? W[o * K + c] : 0.0f;
}

__global__ void k_fold_bn(const float* __restrict__ b,
                          const float* __restrict__ g,
                          const float* __restrict__ be,
                          const float* __restrict__ mu,
                          const float* __restrict__ va,
                          float* __restrict__ s, float* __restrict__ t,
                          int O) {
  int i = blockIdx.x * blockDim.x + threadIdx.x;
  if (i >= O) return;
  float sc = g[i] * rsqrtf(va[i] + 1e-5f);
  s[i] = sc;
  t[i] = (b[i] - mu[i]) * sc + be[i];
}

// ------------------------------------------------------- farthest point sample
// One block per batch; LDS-resident running-min distance array; argmax with
// first-occurrence tie-break to match jnp.argmax.
__global__ void k_fps(const float* __restrict__ xyz, int N, int npoint,
                      int* __restrict__ out) {
  const int b = blockIdx.x;
  const float* p = xyz + (size_t)b * N * 3;
  int* o = out + (size_t)b * npoint;
  __shared__ float dist[4096];
  __shared__ float rv[256];
  __shared__ int ri[256];
  __shared__ float cen[3];
  __shared__ int s_far;
  const int tid = threadIdx.x;
  for (int n = tid; n < N; n += blockDim.x) dist[n] = 1e10f;
  if (tid == 0) s_far = 0;
  __syncthreads();
  for (int it = 0; it < npoint; ++it) {
    if (tid == 0) {
      int far = s_far;
      o[it] = far;
      cen[0] = p[far * 3 + 0];
      cen[1] = p[far * 3 + 1];
      cen[2] = p[far * 3 + 2];
    }
    __syncthreads();
    const float cx = cen[0], cy = cen[1], cz = cen[2];
    float bv = -1.0f;
    int bi = 0;
    for (int n = tid; n < N; n += blockDim.x) {
      float dx = p[n * 3 + 0] - cx;
      float dy = p[n * 3 + 1] - cy;
      float dz = p[n * 3 + 2] - cz;
      float d = dx * dx + dy * dy + dz * dz;
      float dn = fminf(dist[n], d);
      dist[n] = dn;
      if (dn > bv) { bv = dn; bi = n; }
    }
    rv[tid] = bv;
    ri[tid] = bi;
    __syncthreads();
    for (int s = blockDim.x >> 1; s > 0; s >>= 1) {
      if (tid < s) {
        float ov = rv[tid + s];
        int oi = ri[tid + s];
        if (ov > rv[tid] || (ov == rv[tid] && oi < ri[tid])) {
          rv[tid] = ov;
          ri[tid] = oi;
        }
      }
      __syncthreads();
    }
    if (tid == 0) s_far = ri[0];
    __syncthreads();
  }
}

// ------------------------------------------------------------- gather centers
__global__ void k_gather_centers(const float* __restrict__ xyz,
                                 const int* __restrict__ fidx,
                                 float* __restrict__ nxyz, int N, int S,
                                 int BS) {
  int i = blockIdx.x * blockDim.x + threadIdx.x;
  if (i >= BS) return;
  int b = i / S;
  int j = fidx[i];
  const float* src = xyz + ((size_t)b * N + j) * 3;
  nxyz[i * 3 + 0] = src[0];
  nxyz[i * 3 + 1] = src[1];
  nxyz[i * 3 + 2] = src[2];
}

// ------------------------------------------------------------------ ball query
// One wave per query center: wave32 ballot scan collects the first `nsample`
// in-radius indices in ascending order (matches masked-arange top-k), padding
// with the first hit.
__global__ void k_ball(const float* __restrict__ xyz,
                       const float* __restrict__ nxyz, int* __restrict__ idx,
                       float r2, int N, int S, int BS, int nsample) {
  int w = (blockIdx.x * blockDim.x + threadIdx.x) >> 5;
  if (w >= BS) return;
  int lane = threadIdx.x & 31;
  int b = w / S;
  const float* p = xyz + (size_t)b * N * 3;
  float cx = nxyz[w * 3 + 0], cy = nxyz[w * 3 + 1], cz = nxyz[w * 3 + 2];
  int* o = idx + (size_t)w * nsample;
  int count = 0;
  int first = -1;
  for (int base = 0; base < N && count < nsample; base += 32) {
    int n = base + lane;
    bool ok = false;
    if (n < N) {
      float dx = p[n * 3 + 0] - cx;
      float dy = p[n * 3 + 1] - cy;
      float dz = p[n * 3 + 2] - cz;
      ok = (dx * dx + dy * dy + dz * dz) <= r2;
    }
    unsigned m = (unsigned)__ballot(ok);
    while (m && count < nsample) {
      int t = __ffs(m) - 1;
      m &= m - 1;
      if (first < 0) first = base + t;
      if (lane == 0) o[count] = base + t;
      count++;
    }
  }
  if (lane == 0)
    for (int i2 = count; i2 < nsample; i2++) o[i2] = (first < 0 ? 0 : first);
}

// ----------------------------------------------- fused SA: gather+MLPx3+maxpool
struct SAArgs {
  const float* W[3];  // K-padded weights, row stride Kp[l]
  const float* s[3];  // folded BN scale
  const float* t[3];  // folded BN shift
  int Kp[3];          // padded input channels (stride), multiple of 4
  int O[3];           // output channels per layer (multiples of 16)
};

// One block (128 threads = 4 waves) per ball-group. The 16 samples of the
// group are one WMMA M-tile; activations ping-pong through LDS.
__global__ void k_sa_mlp(const float* __restrict__ xyz,
                         const float* __restrict__ pts, int C,
                         const float* __restrict__ nxyz,
                         const int* __restrict__ idx,
                         float* __restrict__ outFeat, int N, int S, SAArgs A) {
  __shared__ float buf0[16 * 144];  // layer0 input (Kp<=132) / layer1 output
  __shared__ float buf1[16 * 256];  // layer0 output / layer2 output
  const int g = blockIdx.x;  // b*S + s
  const int b = g / S;
  const int tid = threadIdx.x;
  const int Kin = 3 + C;
  const int Kp = A.Kp[0];
  const int* id = idx + (size_t)g * 16;
  const float c0 = nxyz[g * 3 + 0], c1 = nxyz[g * 3 + 1], c2 = nxyz[g * 3 + 2];
  // Gather grouped (center-relative xyz ++ point feats), zero-pad K to Kp.
  for (int i = tid; i < 16 * Kp; i += blockDim.x) {
    int k = i / Kp, c = i % Kp;
    float v = 0.0f;
    int j = id[k];
    size_t base = (size_t)b * N + j;
    if (c < 3)
      v = xyz[base * 3 + c] - (c == 0 ? c0 : (c == 1 ? c1 : c2));
    else if (c < Kin)
      v = pts[base * C + (c - 3)];
    buf0[k * Kp + c] = v;
  }
  __syncthreads();
  const int lane = tid & 31;
  const int wv = tid >> 5;  // 0..3
  const int l15 = lane & 15;
  const int hi = lane >> 4;  // 0 or 1
  float* cur = buf0;
  float* nxt = buf1;
  int stride = Kp;
  for (int l = 0; l < 3; l++) {
    const int Ks = stride;  // == weight row stride (padded)
    const int O = A.O[l];
    const float* Wl = A.W[l];
    const int T = O >> 4;
    const float* ar = cur + l15 * Ks + hi * 2;  // A operand base for this lane
    for (int t = wv; t < T; t += 4) {
      v8f acc = {};
      const int col = t * 16 + l15;
      const float* wr = Wl + (size_t)col * Ks + hi * 2;
      for (int k0 = 0; k0 < Ks; k0 += 4) {
        v2f a = *(const v2f*)(ar + k0);
        v2f bv = *(const v2f*)(wr + k0);
        acc = wmma4(a, bv, acc);
      }
      // folded BatchNorm + ReLU epilogue (per-output-channel scale/shift)
      const float sc = A.s[l][col];
      const float tt = A.t[l][col];
#pragma unroll
      for (int i = 0; i < 8; i++) {
        const int m = i + hi * 8;
        nxt[m * O + col] = fmaxf(acc[i] * sc + tt, 0.0f);
      }
    }
    __syncthreads();
    float* tmp = cur;
    cur = nxt;
    nxt = tmp;
    stride = O;
  }
  // max over the 16 samples
  const int O2 = A.O[2];
  for (int c = tid; c < O2; c += blockDim.x) {
    float m = cur[c];
    for (int k = 1; k < 16; k++) m = fmaxf(m, cur[k * O2 + c]);
    outFeat[(size_t)g * O2 + c] = m;
  }
}

// ---------------------------------------------- generic WMMA linear+BN+ReLU
// One wave per 16x16 output tile. Requires R%16==0 and K%4==0 (all callers).
__global__ void k_linear_bn_relu(const float* __restrict__ X,
                                 const float* __restrict__ W,
                                 const float* __restrict__ sA,
                                 const float* __restrict__ tA,
                                 float* __restrict__ Y, int R, int K, int O) {
  const int w = (blockIdx.x * blockDim.x + threadIdx.x) >> 5;
  const int CT = O >> 4;
  if (w >= (R >> 4) * CT) return;  // wave-uniform exit (EXEC stays all-ones)
  const int tr = w / CT, tc = w % CT;
  const int lane = threadIdx.x & 31;
  const int l15 = lane & 15, hi = lane >> 4;
  const float* xr = X + (size_t)(tr * 16 + l15) * K + hi * 2;
  const int col = tc * 16 + l15;
  const float* wr = W + (size_t)col * K + hi * 2;
  v8f acc = {};
  for (int k0 = 0; k0 < K; k0 += 4) {
    v2f a = *(const v2f*)(xr + k0);
    v2f bv = *(const v2f*)(wr + k0);
    acc = wmma4(a, bv, acc);
  }
  const float sc = sA[col];
  const float tt = tA[col];
  float* yb = Y + (size_t)(tr * 16) * O + tc * 16 + l15;
#pragma unroll
  for (int i = 0; i < 8; i++) {
    const int m = i + hi * 8;
    yb[(size_t)m * O] = fmaxf(acc[i] * sc + tt, 0.0f);
  }
}

// ------------------------------------------- 3-NN inverse-distance interpolate
// One thread per upsampled point; low-res xyz staged in LDS. Writes the
// concatenated [points1 ++ interp] row for the following FP MLP.
__global__ void k_fp_interp(const float* __restrict__ xyz1,
                            const float* __restrict__ xyz2,
                            const float* __restrict__ p1, int C1,
                            const float* __restrict__ p2, int C2,
                            float* __restrict__ out, int N1, int N2) {
  __shared__ float s2[1184 * 3];
  const int b = blockIdx.y;
  const float* x2 = xyz2 + (size_t)b * N2 * 3;
  for (int i = threadIdx.x; i < N2 * 3; i += blockDim.x) s2[i] = x2[i];
  __syncthreads();
  const int n = blockIdx.x * blockDim.x + threadIdx.x;
  if (n >= N1) return;
  const size_t r = (size_t)b * N1 + n;
  const float cx = xyz1[r * 3 + 0], cy = xyz1[r * 3 + 1], cz = xyz1[r * 3 + 2];
  float d0 = 3.4e38f, d1 = 3.4e38f, d2 = 3.4e38f;
  int i0 = 0, i1 = 0, i2 = 0;
  for (int j = 0; j < N2; j++) {
    float dx = s2[j * 3 + 0] - cx;
    float dy = s2[j * 3 + 1] - cy;
    float dz = s2[j * 3 + 2] - cz;
    float d = dx * dx + dy * dy + dz * dz;
    if (d < d0) { d2 = d1; i2 = i1; d1 = d0; i1 = i0; d0 = d; i0 = j; }
    else if (d < d1) { d2 = d1; i2 = i1; d1 = d; i1 = j; }
    else if (d < d2) { d2 = d; i2 = j; }
  }
  float w0 = 1.0f / (d0 + 1e-8f);
  float w1 = 1.0f / (d1 + 1e-8f);
  float w2 = 1.0f / (d2 + 1e-8f);
  const float ws = w0 + w1 + w2;
  w0 /= ws; w1 /= ws; w2 /= ws;
  const int CO = C1 + C2;
  float* o = out + r * CO;
  if (p1) {
    const float* q = p1 + r * C1;
    for (int c = 0; c < C1; c++) o[c] = q[c];
  }
  const float* q0 = p2 + ((size_t)b * N2 + i0) * C2;
  const float* q1 = p2 + ((size_t)b * N2 + i1) * C2;
  const float* q2 = p2 + ((size_t)b * N2 + i2) * C2;
  for (int c = 0; c < C2; c++)
    o[C1 + c] = w0 * q0[c] + w1 * q1[c] + w2 * q2[c];
}

// ---------------------------------------------------------------- final 1-ch
__global__ void k_conv2(const float* __restrict__ X, const float* __restrict__ W,
                        const float* __restrict__ b, float* __restrict__ out,
                        int R, int K) {
  int i = blockIdx.x * blockDim.x + threadIdx.x;
  if (i >= R) return;
  const float* x = X + (size_t)i * K;
  float acc = b[0];
  for (int c = 0; c < K; c++) acc += x[c] * W[c];
  out[i] = acc;
}

// ===========================================================================
struct Layer {
  const float* W;
  const float* b;
  const float* gamma;
  const float* beta;
  const float* mean;
  const float* var;
  int cin, cout;
};

struct Prep {
  const float* W;  // possibly K-padded
  const float* s;
  const float* t;
  int Kp, O;
};

extern "C" void kernel_launch(void* const* d_in, const int* in_sizes, int n_in,
                              void* d_out, int out_size, void* d_ws,
                              size_t ws_size, hipStream_t stream) {
  (void)in_sizes; (void)n_in; (void)out_size; (void)ws_size;
  const int B = 16, N0 = 4096;
  const int S1 = 1180, S2 = 295, S3 = 59, NS = 16;

  // ---- input unpack (setup_inputs() insertion order) ----
  const float* pc = (const float*)d_in[0];
  int p = 1;
  auto take = [&](int cin, int cout) {
    Layer l;
    l.W = (const float*)d_in[p++];
    l.b = (const float*)d_in[p++];
    l.gamma = (const float*)d_in[p++];
    l.beta = (const float*)d_in[p++];
    l.mean = (const float*)d_in[p++];
    l.var = (const float*)d_in[p++];
    l.cin = cin; l.cout = cout;
    return l;
  };
  Layer sa1[3] = {take(4, 32), take(32, 32), take(32, 64)};
  Layer sa2[3] = {take(67, 64), take(64, 64), take(64, 128)};
  Layer sa3[3] = {take(131, 128), take(128, 128), take(128, 256)};
  Layer fp3[2] = {take(384, 256), take(256, 256)};
  Layer fp2[2] = {take(320, 256), take(256, 128)};
  Layer fp1[3] = {take(128, 128), take(128, 128), take(128, 128)};
  Layer head1 = take(128, 128);
  const float* conv2_W = (const float*)d_in[p++];
  const float* conv2_b = (const float*)d_in[p++];

  // ---- workspace carving ----
  char* base = (char*)d_ws;
  size_t off = 0;
  auto alloc = [&](size_t bytes) -> void* {
    off = (off + 255) & ~(size_t)255;
    void* r = base + off;
    off += bytes;
    return r;
  };
  float* xyz0 = (float*)alloc((size_t)B * N0 * 3 * 4);
  float* feat0 = (float*)alloc((size_t)B * N0 * 4);
  int* fps1 = (int*)alloc((size_t)B * S1 * 4);
  float* nxyz1 = (float*)alloc((size_t)B * S1 * 3 * 4);
  int* idx1 = (int*)alloc((size_t)B * S1 * NS * 4);
  float* l1p = (float*)alloc((size_t)B * S1 * 64 * 4);
  int* fps2 = (int*)alloc((size_t)B * S2 * 4);
  float* nxyz2 = (float*)alloc((size_t)B * S2 * 3 * 4);
  int* idx2 = (int*)alloc((size_t)B * S2 * NS * 4);
  float* l2p = (float*)alloc((size_t)B * S2 * 128 * 4);
  int* fps3 = (int*)alloc((size_t)B * S3 * 4);
  float* nxyz3 = (float*)alloc((size_t)B * S3 * 3 * 4);
  int* idx3 = (int*)alloc((size_t)B * S3 * NS * 4);
  float* l3p = (float*)alloc((size_t)B * S3 * 256 * 4);
  float* l2pn = (float*)alloc((size_t)B * S2 * 256 * 4);
  float* l1pn = (float*)alloc((size_t)B * S1 * 128 * 4);

  // ---- prep: pad odd-K weights, fold BN (runs every call; deterministic) ---
  auto prep = [&](const Layer& L) -> Prep {
    Prep P;
    P.O = L.cout;
    int Kp = (L.cin + 3) & ~3;
    P.Kp = Kp;
    if (Kp != L.cin) {
      float* Wp = (float*)alloc((size_t)L.cout * Kp * 4);
      k_pack_w<<<(L.cout * Kp + 255) / 256, 256, 0, stream>>>(L.W, Wp, L.cout,
                                                              L.cin, Kp);
      P.W = Wp;
    } else {
      P.W = L.W;
    }
    float* s = (float*)alloc((size_t)L.cout * 4);
    float* t = (float*)alloc((size_t)L.cout * 4);
    k_fold_bn<<<(L.cout + 255) / 256, 256, 0, stream>>>(
        L.b, L.gamma, L.beta, L.mean, L.var, s, t, L.cout);
    P.s = s;
    P.t = t;
    return P;
  };
  Prep psa1[3] = {prep(sa1[0]), prep(sa1[1]), prep(sa1[2])};
  Prep psa2[3] = {prep(sa2[0]), prep(sa2[1]), prep(sa2[2])};
  Prep psa3[3] = {prep(sa3[0]), prep(sa3[1]), prep(sa3[2])};
  Prep pfp3[2] = {prep(fp3[0]), prep(fp3[1])};
  Prep pfp2[2] = {prep(fp2[0]), prep(fp2[1])};
  Prep pfp1[3] = {prep(fp1[0]), prep(fp1[1]), prep(fp1[2])};
  Prep phead = prep(head1);

  const size_t mark = off;  // per-stage scratch reused below (stream-serial)

  auto mkSA = [&](const Prep* ls) {
    SAArgs a;
    for (int i = 0; i < 3; i++) {
      a.W[i] = ls[i].W;
      a.s[i] = ls[i].s;
      a.t[i] = ls[i].t;
      a.Kp[i] = ls[i].Kp;
      a.O[i] = ls[i].O;
    }
    return a;
  };
  auto lin = [&](const float* X, const Prep& ly, float* Y, int R) {
    int waves = (R / 16) * (ly.O / 16);
    int blocks = (waves * 32 + 255) / 256;
    k_linear_bn_relu<<<blocks, 256, 0, stream>>>(X, ly.W, ly.s, ly.t, Y, R,
                                                 ly.Kp, ly.O);
  };

  // ---- stage 0: split (B,N,4) -> xyz (B,N,3), feats (B,N,1) ----
  k_split<<<(B * N0 + 255) / 256, 256, 0, stream>>>(pc, xyz0, feat0, B * N0);

  // ---- SA1 ----
  k_fps<<<B, 256, 0, stream>>>(xyz0, N0, S1, fps1);
  k_gather_centers<<<(B * S1 + 255) / 256, 256, 0, stream>>>(xyz0, fps1, nxyz1,
                                                             N0, S1, B * S1);
  k_ball<<<(B * S1 * 32 + 255) / 256, 256, 0, stream>>>(
      xyz0, nxyz1, idx1, 0.2f * 0.2f, N0, S1, B * S1, NS);
  k_sa_mlp<<<B * S1, 128, 0, stream>>>(xyz0, feat0, 1, nxyz1, idx1, l1p, N0,
                                       S1, mkSA(psa1));

  // ---- SA2 ----
  k_fps<<<B, 256, 0, stream>>>(nxyz1, S1, S2, fps2);
  k_gather_centers<<<(B * S2 + 255) / 256, 256, 0, stream>>>(nxyz1, fps2,
                                                             nxyz2, S1, S2,
                                                             B * S2);
  k_ball<<<(B * S2 * 32 + 255) / 256, 256, 0, stream>>>(
      nxyz1, nxyz2, idx2, 0.4f * 0.4f, S1, S2, B * S2, NS);
  k_sa_mlp<<<B * S2, 128, 0, stream>>>(nxyz1, l1p, 64, nxyz2, idx2, l2p, S1,
                                       S2, mkSA(psa2));

  // ---- SA3 ----
  k_fps<<<B, 256, 0, stream>>>(nxyz2, S2, S3, fps3);
  k_gather_centers<<<(B * S3 + 255) / 256, 256, 0, stream>>>(nxyz2, fps3,
                                                             nxyz3, S2, S3,
                                                             B * S3);
  k_ball<<<(B * S3 * 32 + 255) / 256, 256, 0, stream>>>(
      nxyz2, nxyz3, idx3, 0.6f * 0.6f, S2, S3, B * S3, NS);
  k_sa_mlp<<<B * S3, 128, 0, stream>>>(nxyz2, l2p, 128, nxyz3, idx3, l3p, S2,
                                       S3, mkSA(psa3));

  // ---- FP3: l2 <- interp(l3) ----
  {
    off = mark;
    float* cat = (float*)alloc((size_t)B * S2 * 384 * 4);
    float* mid = (float*)alloc((size_t)B * S2 * 256 * 4);
    k_fp_interp<<<dim3((S2 + 255) / 256, B), 256, 0, stream>>>(
        nxyz2, nxyz3, l2p, 128, l3p, 256, cat, S2, S3);
    lin(cat, pfp3[0], mid, B * S2);
    lin(mid, pfp3[1], l2pn, B * S2);
  }

  // ---- FP2: l1 <- interp(l2) ----
  {
    off = mark;
    float* cat = (float*)alloc((size_t)B * S1 * 320 * 4);
    float* mid = (float*)alloc((size_t)B * S1 * 256 * 4);
    k_fp_interp<<<dim3((S1 + 255) / 256, B), 256, 0, stream>>>(
        nxyz1, nxyz2, l1p, 64, l2pn, 256, cat, S1, S2);
    lin(cat, pfp2[0], mid, B * S1);
    lin(mid, pfp2[1], l1pn, B * S1);
  }

  // ---- FP1 + head + conv2 ----
  {
    off = mark;
    float* cat = (float*)alloc((size_t)B * N0 * 128 * 4);
    float* t1 = (float*)alloc((size_t)B * N0 * 128 * 4);
    k_fp_interp<<<dim3((N0 + 255) / 256, B), 256, 0, stream>>>(
        xyz0, nxyz1, nullptr, 0, l1pn, 128, cat, N0, S1);
    lin(cat, pfp1[0], t1, B * N0);
    lin(t1, pfp1[1], cat, B * N0);
    lin(cat, pfp1[2], t1, B * N0);
    lin(t1, phead, cat, B * N0);
    k_conv2<<<(B * N0 + 255) / 256, 256, 0, stream>>>(cat, conv2_W, conv2_b,
                                                      (float*)d_out, B * N0,
                                                      128);
  }
}